// MMDGLM_420906795741
// MI455X (gfx1250) — compile-verified
//
#include <hip/hip_runtime.h>
#include <math.h>

#define T_DIM   2000
#define KFR     400
#define KD      400
#define A_DIM   100
#define M_ROWS  (T_DIM * KFR)        // 800000 rows of X (length A_DIM)
#define G_GROUPS (M_ROWS / 4)        // 200000 groups of 4 rows
#define NB      512                  // blocks in main kernel
#define WPB     8                    // waves per block (256 threads, wave32)
#define TOTAL_WAVES (NB * WPB)
#define ROW_STEP (4 * TOTAL_WAVES)   // 16384 rows advanced per loop iteration
#define K_STEP  (ROW_STEP % KFR)     // 384
#define EPSF    1e-24f

// d_ws float layout
#define WS_PART_STRIDE 128                     // per-block partial: [0..111]=grad(a padded), [112]=scalar
#define WS_COEF (NB * WS_PART_STRIDE)          // c[k], 400 floats
#define WS_DT   (WS_COEF + KFR)
#define WS_MMD  (WS_DT + 1)

typedef float v2f __attribute__((ext_vector_type(2)));
typedef float v8f __attribute__((ext_vector_type(8)));

// ---------------- prologue: phi reductions -> coef[k], dt, closed-form mmd ----------------
__global__ void mmdglm_prologue(const float* __restrict__ t,
                                const float* __restrict__ phi_d,
                                const float* __restrict__ phi_fr,
                                float* __restrict__ ws) {
    __shared__ float buf[512];
    __shared__ float sums[4];
    const int tid = threadIdx.x;
    float pf = (tid < KFR) ? phi_fr[tid] : 0.f;
    float pd = (tid < KD)  ? phi_d[tid]  : 0.f;
    float vals[4] = {pf, pf * pf, pd, pd * pd};
    for (int i = 0; i < 4; ++i) {
        buf[tid] = vals[i];
        __syncthreads();
        for (int s = 256; s >= 1; s >>= 1) {
            if (tid < s) buf[tid] += buf[tid + s];
            __syncthreads();
        }
        if (tid == 0) sums[i] = buf[0];
        __syncthreads();
    }
    const float S_fr = sums[0], Ssq_fr = sums[1], S_d = sums[2], Ssq_d = sums[3];
    const float n_fr = (float)(KFR * (KFR - 1) / 2);
    const float n_d  = (float)(KD * (KD - 1) / 2);
    if (tid < KFR) {
        // same coefficient serves both mmd_grad (on ll) and grad_theta (on g_ll)
        float c = (pf * S_fr - pf * pf) / n_fr - 2.f * S_d * pf / ((float)KD * (float)KFR);
        ws[WS_COEF + tid] = c;
    }
    if (tid == 0) {
        ws[WS_DT] = t[1] - t[0];
        float tri_ff = (S_fr * S_fr - Ssq_fr) * 0.5f / n_fr;
        float tri_dd = (S_d * S_d - Ssq_d) * 0.5f / n_d;
        ws[WS_MMD] = tri_dd + tri_ff - 2.f * S_d * S_fr / ((float)KD * (float)KFR);
    }
}

// ---------------- main: fused single pass over X with f32 WMMA accumulation ----------------
__global__ __launch_bounds__(256) void mmdglm_main(const float* __restrict__ X,
                                                   const int* __restrict__ mask,
                                                   const float* __restrict__ theta,
                                                   float* __restrict__ ws) {
    const int tid    = threadIdx.x;
    const int lane   = tid & 31;
    const int wid    = tid >> 5;
    const int lane16 = lane & 15;
    const int hi     = lane >> 4;          // 0: lanes 0-15, 1: lanes 16-31
    const int a6     = 96 + (lane16 & 3);  // always-in-bounds address for padded chunk 6
    const bool ok6   = (lane16 < 4);

    const float* __restrict__ coef = ws + WS_COEF;
    const float dt = ws[WS_DT];

    // per-lane theta slice: th[c] = theta[16c + lane16], zero-padded past A_DIM
    float th[7];
#pragma unroll
    for (int c = 0; c < 6; ++c) th[c] = theta[16 * c + lane16];
    th[6] = ok6 ? theta[a6] : 0.f;

    const v8f vzero = {0.f, 0.f, 0.f, 0.f, 0.f, 0.f, 0.f, 0.f};
    v8f acc[7];
#pragma unroll
    for (int c = 0; c < 7; ++c) acc[c] = vzero;
    float s_acc = 0.f;

    const int waveG = blockIdx.x * WPB + wid;
    const int rStart = 4 * waveG + hi * 2;           // this half-wave's even row
    const float* pA = X + (size_t)rStart * A_DIM;    // row rA; row rB = pA + A_DIM
    const int*   pM = mask + rStart;
    int k0 = (4 * waveG) % KFR;                      // trial index of r0, maintained incrementally

    for (int g = waveG; g < G_GROUPS; g += TOTAL_WAVES) {
        // A-matrix tiles (16x4 f32 layout): coalesced 16-float row segments per half-wave
        float t0[7], t1[7];
#pragma unroll
        for (int c = 0; c < 6; ++c) {
            int a = 16 * c + lane16;
            t0[c] = pA[a];
            t1[c] = pA[A_DIM + a];
        }
        {   // padded chunk: unconditional clamped load, then zero via cndmask
            float x0 = pA[a6];
            float x1 = pA[A_DIM + a6];
            t0[6] = ok6 ? x0 : 0.f;
            t1[6] = ok6 ? x1 : 0.f;
        }

        // u = X·theta from the same registers (per-lane partial + 16-lane xor reduce)
        float p0 = 0.f, p1 = 0.f;
#pragma unroll
        for (int c = 0; c < 7; ++c) {
            p0 = fmaf(t0[c], th[c], p0);
            p1 = fmaf(t1[c], th[c], p1);
        }
#pragma unroll
        for (int m = 1; m < 16; m <<= 1) {
            p0 += __shfl_xor(p0, m, 32);
            p1 += __shfl_xor(p1, m, 32);
        }
        // p0 = u[rA] replicated in this half; p1 = u[rB]

        const int   mA = pM[0];
        const int   mB = pM[1];
        const int   kA = k0 + hi * 2;                // k0 is a multiple of 4, so no wrap
        const float cA = coef[kA];
        const float cB = coef[kA + 1];

        float rrA = __expf(p0);
        float rrB = __expf(p1);
        float erA = __expf(dt * rrA), enA = __expf(-dt * rrA);
        float erB = __expf(dt * rrB), enB = __expf(-dt * rrB);
        float wA  = mA ? (rrA / (erA - 1.f - EPSF)) : (-rrA);
        float wB  = mB ? (rrB / (erB - 1.f - EPSF)) : (-rrB);
        float llA = mA ? __logf(1.f - enA + EPSF) : (-dt * rrA);
        float llB = mB ? __logf(1.f - enB + EPSF) : (-dt * rrB);

        s_acc += cA * llA + cB * llB;        // replicated 16x per half; divided later
        float vA = dt * cA * wA;             // B-matrix value for rows r0 / r0+2
        float vB = dt * cB * wB;             // B-matrix value for rows r0+1 / r0+3

        v2f bvec = {vA, vB};
#pragma unroll
        for (int c = 0; c < 7; ++c) {
            v2f avec = {t0[c], t1[c]};
            // D[a_local, n] += sum_kk X[r0+kk][a0+a_local] * v[r0+kk]
            acc[c] = __builtin_amdgcn_wmma_f32_16x16x4_f32(
                false, avec, false, bvec, (short)0, acc[c], false, false);
        }

        // incremental advance (no div/mod, no 64-bit mad in the loop)
        pA += (size_t)ROW_STEP * A_DIM;
        pM += ROW_STEP;
        k0 += K_STEP;
        if (k0 >= KFR) k0 -= KFR;
    }

    // ---- deterministic block reduction ----
    __shared__ float sred[WPB][WS_PART_STRIDE];
#pragma unroll
    for (int m = 1; m < 32; m <<= 1) s_acc += __shfl_xor(s_acc, m, 32);
    if (lane == 0) sred[wid][112] = s_acc * (1.f / 16.f);
#pragma unroll
    for (int c = 0; c < 7; ++c) {
#pragma unroll
        for (int q = 0; q < 8; ++q) {
            float val = acc[c][q];           // lanes 0-15: a=16c+q ; lanes 16-31: a=16c+q+8
            if (lane16 == 0) sred[wid][16 * c + q + 8 * hi] = val;
        }
    }
    __syncthreads();
    if (tid < 113) {
        float sum = 0.f;
#pragma unroll
        for (int w = 0; w < WPB; ++w) sum += sred[w][tid];
        ws[blockIdx.x * WS_PART_STRIDE + tid] = sum;
    }
}

// ---------------- final: fixed-order sum over block partials ----------------
__global__ void mmdglm_final(const float* __restrict__ ws, float* __restrict__ out) {
    const int tid = threadIdx.x;
    if (tid < A_DIM) {
        float s = 0.f;
        for (int b = 0; b < NB; ++b) s += ws[b * WS_PART_STRIDE + tid];
        out[1 + tid] = s;                    // grad_theta[a]
    } else if (tid == A_DIM) {
        float s = 0.f;
        for (int b = 0; b < NB; ++b) s += ws[b * WS_PART_STRIDE + 112];
        out[0] = s;                          // mmd_grad
    } else if (tid == A_DIM + 1) {
        out[A_DIM + 1] = ws[WS_MMD];         // mmd (closed form)
    }
}

extern "C" void kernel_launch(void* const* d_in, const int* in_sizes, int n_in,
                              void* d_out, int out_size, void* d_ws, size_t ws_size,
                              hipStream_t stream) {
    const float* t      = (const float*)d_in[0];
    const int*   mask   = (const int*)d_in[1];
    const float* X      = (const float*)d_in[2];
    const float* theta  = (const float*)d_in[3];
    const float* phi_d  = (const float*)d_in[4];
    const float* phi_fr = (const float*)d_in[5];
    float* out = (float*)d_out;
    float* ws  = (float*)d_ws;

    mmdglm_prologue<<<1, 512, 0, stream>>>(t, phi_d, phi_fr, ws);
    mmdglm_main<<<NB, 256, 0, stream>>>(X, mask, theta, ws);
    mmdglm_final<<<1, 128, 0, stream>>>(ws, out);
}